// PatchShuffle_88227218194453
// MI455X (gfx1250) — compile-verified
//
#include <hip/hip_runtime.h>
#include <hip/hip_bf16.h>

// PatchShuffle for MI455X (gfx1250, wave32).
// T=196, B=256, C=768, remain_T=49.
// Memory-bound (~620 MB traffic -> ~27us @ 23.3 TB/s). Distance row-sums are
// done with V_WMMA_F32_16X16X4_F32 (exact f32) so argsort ordering matches
// the f32 reference.

typedef __attribute__((ext_vector_type(2))) float v2f;
typedef __attribute__((ext_vector_type(8))) float v8f;

#define NT 196
#define NB 256
#define NC 768
#define NREMAIN 49
#define NROWS (NT * NB)          // 50176 (b*T + t) rows, fully contiguous in C

// ---------------------------------------------------------------------------
// Kernel 1: pam_dist[r] = sum_c |shift[r*768+c] - mean[r*768+c]|
// One wave handles 16 rows. Per ISA A-matrix 16x4 f32 layout:
//   lanes 0-15  (M = lane)    hold {K=0, K=1}  -> chunks [0,192) and [192,384)
//   lanes 16-31 (M = lane-16) hold {K=2, K=3}  -> chunks [384,576) and [576,768)
// WMMA with all-ones B broadcasts exact f32 row sums into C columns.
// ---------------------------------------------------------------------------
__global__ __launch_bounds__(256) void pam_dist_kernel(
    const float* __restrict__ shift,
    const float* __restrict__ mean,
    float* __restrict__ dist)
{
    const int lane    = threadIdx.x & 31;
    const int wave    = threadIdx.x >> 5;
    const int waveIdx = blockIdx.x * 8 + wave;   // 3136 waves total
    const int rowBase = waveIdx * 16;
    const int row     = rowBase + (lane & 15);

    const size_t base = (size_t)row * NC + ((lane & 16) ? 384 : 0);
    const float4* __restrict__ s = (const float4*)(shift + base);
    const float4* __restrict__ m = (const float4*)(mean  + base);

    float sx = 0.0f, sy = 0.0f;
#pragma unroll 4
    for (int i = 0; i < 48; ++i) {           // 192 floats per chunk = 48 float4
        float4 a0 = s[i];
        float4 b0 = m[i];
        sx += fabsf(a0.x - b0.x) + fabsf(a0.y - b0.y) +
              fabsf(a0.z - b0.z) + fabsf(a0.w - b0.w);
        float4 a1 = s[i + 48];
        float4 b1 = m[i + 48];
        sy += fabsf(a1.x - b1.x) + fabsf(a1.y - b1.y) +
              fabsf(a1.z - b1.z) + fabsf(a1.w - b1.w);
    }

    v2f A;  A.x = sx;   A.y = sy;
    v2f Bv; Bv.x = 1.0f; Bv.y = 1.0f;        // 4x16 all-ones B
    v8f C = {};
    // D[M,N] = sum_K A[M,K] * 1 + 0  (exact f32 row sums, broadcast over N)
    C = __builtin_amdgcn_wmma_f32_16x16x4_f32(
            /*neg_a=*/false, A, /*neg_b=*/false, Bv,
            /*c_mod=*/(short)0, C, /*reuse_a=*/false, /*reuse_b=*/false);

    // C layout: VGPR i -> lanes 0-15: M=i (N=lane), lanes 16-31: M=8+i.
    // Lane 0 (N=0) holds sums of rows rowBase+0..7; lane 16 rows rowBase+8..15.
    if ((lane & 15) == 0) {
        const int mb = rowBase + ((lane >> 4) << 3);
#pragma unroll
        for (int i = 0; i < 8; ++i) dist[mb + i] = C[i];
    }
}

// ---------------------------------------------------------------------------
// Kernel 2: stable argsort of 196 dists per batch via rank counting.
// rank(t) = #{j : d[j] < d[t]} + #{j < t : d[j] == d[t]}   (stable, matches
// jnp.argsort). Writes order_ws (int) plus forward/backward index outputs.
// ---------------------------------------------------------------------------
__global__ __launch_bounds__(256) void argsort_kernel(
    const float* __restrict__ dist,
    int* __restrict__ order_ws,
    float* __restrict__ out)
{
    __shared__ float d[NT];
    const int b = blockIdx.x;
    const int t = threadIdx.x;

    if (t < NT) d[t] = dist[b * NT + t];
    __syncthreads();

    if (t < NT) {
        const float dt = d[t];
        int rank = 0;
#pragma unroll 4
        for (int j = 0; j < NT; ++j) {
            const float dj = d[j];
            rank += (dj < dt) || ((dj == dt) && (j < t));
        }
        order_ws[b * NT + rank] = t;

        const size_t FWD = (size_t)NT * NB * NC;            // 38,535,168
        const size_t BWD = FWD + (size_t)NREMAIN * NB;       // + 12,544
        if (rank < NREMAIN)
            out[FWD + (size_t)rank * NB + b] = (float)t;     // asforward[k,b]
        out[BWD + (size_t)rank * NB + b] = (float)t;         // asbackward[k,b]
    }
}

// ---------------------------------------------------------------------------
// Kernel 3: gather. out[k,b,:] = patches[order[b,k], b, :].
// useful (k<49) and mask (k>=49) regions are contiguous in the flat output,
// so one pass over r = k*256 + b covers both. One wave per 3KB row:
// 6x float4 per lane -> coalesced global_load_b128 / global_store_b128.
// ---------------------------------------------------------------------------
__global__ __launch_bounds__(256) void gather_kernel(
    const float* __restrict__ patches,
    const int* __restrict__ order_ws,
    float* __restrict__ out)
{
    const int lane = threadIdx.x & 31;
    const int wave = threadIdx.x >> 5;
    const int r    = blockIdx.x * 8 + wave;   // [0, 50176)
    const int k    = r >> 8;                  // output slot 0..195
    const int b    = r & 255;

    const int src_t = order_ws[b * NT + k];

    const float4* __restrict__ src =
        (const float4*)(patches + ((size_t)src_t * NB + b) * NC);
    float4* __restrict__ dst = (float4*)(out + (size_t)r * NC);

#pragma unroll
    for (int i = 0; i < 6; ++i) {
        dst[lane + i * 32] = src[lane + i * 32];
    }
}

// ---------------------------------------------------------------------------
extern "C" void kernel_launch(void* const* d_in, const int* in_sizes, int n_in,
                              void* d_out, int out_size, void* d_ws, size_t ws_size,
                              hipStream_t stream) {
    const float* patches = (const float*)d_in[0];   // (196, 256, 768) f32
    const float* shift   = (const float*)d_in[1];   // (256, 196, 768) f32
    const float* mean    = (const float*)d_in[2];   // (256, 196, 768) f32
    float* out = (float*)d_out;

    // Workspace layout: [50176 f32 dist][50176 i32 order]  (~400 KB)
    float* dist_ws  = (float*)d_ws;
    int*   order_ws = (int*)((char*)d_ws + (size_t)NROWS * sizeof(float));

    // 3136 waves (16 rows each) / 8 waves per block = 392 blocks
    pam_dist_kernel<<<NROWS / (16 * 8), 256, 0, stream>>>(shift, mean, dist_ws);

    // one block per batch
    argsort_kernel<<<NB, 256, 0, stream>>>(dist_ws, order_ws, out);

    // 50176 row copies / 8 waves per block = 6272 blocks
    gather_kernel<<<NROWS / 8, 256, 0, stream>>>(patches, order_ws, out);
}